// NestedGCN_55946243998144
// MI455X (gfx1250) — compile-verified
//
#include <hip/hip_runtime.h>
#include <hip/hip_bf16.h>
#include <math.h>

#define HID 64

typedef __attribute__((ext_vector_type(2))) float v2f;
typedef __attribute__((ext_vector_type(8))) float v8f;

// ---------------- elementwise / utility kernels ----------------

__global__ void k_fill(float* __restrict__ p, float v, int n) {
    int i = blockIdx.x * blockDim.x + threadIdx.x;
    if (i < n) p[i] = v;
}

__global__ void k_deg(const int* __restrict__ dst, float* __restrict__ deg, int E) {
    int e = blockIdx.x * blockDim.x + threadIdx.x;
    if (e < E) atomicAdd(&deg[dst[e]], 1.0f);
}

__global__ void k_rsqrt(float* __restrict__ d, int n) {
    int i = blockIdx.x * blockDim.x + threadIdx.x;
    if (i < n) d[i] = rsqrtf(d[i]);
}

// layer 1: x is N x 1, W1 is 1 x 64  ->  t[n][j] = x[n] * W1[j]
__global__ void k_layer1(const float* __restrict__ x, const float* __restrict__ W1,
                         float* __restrict__ t, int N) {
    int i = blockIdx.x * blockDim.x + threadIdx.x;
    if (i < N * HID) {
        int n = i >> 6, j = i & 63;
        t[i] = x[n] * W1[j];
    }
}

// hhat = t * dinv[row]; t (== agg buffer) zeroed for the scatter phase.
// NOTE: t_agg is intentionally read-then-written at the same index (aliased reuse).
__global__ void k_hhat_zero(float* t_agg, const float* __restrict__ dinv,
                            float* __restrict__ hhat, int N) {
    int i = blockIdx.x * blockDim.x + threadIdx.x;
    if (i < N * HID) {
        int n = i >> 6;
        hhat[i] = t_agg[i] * dinv[n];
        t_agg[i] = 0.0f;
    }
}

// one 32-lane group per edge; each lane moves a float2 (features 2l, 2l+1)
__global__ void k_scatter(const int* __restrict__ src, const int* __restrict__ dst,
                          const float* __restrict__ hhat, float* agg, int E) {
    long long tid = (long long)blockIdx.x * blockDim.x + threadIdx.x;
    int e = (int)(tid >> 5);
    if (e >= E) return;
    int l = (int)(tid & 31);
    int s = src[e], d = dst[e];
    const float2 v = *(const float2*)(hhat + (size_t)s * HID + 2 * l);
    atomicAdd(&agg[(size_t)d * HID + 2 * l + 0], v.x);
    atomicAdd(&agg[(size_t)d * HID + 2 * l + 1], v.y);
}

// h = dinv * (agg + hhat) + b
__global__ void k_finalize(const float* __restrict__ agg, const float* __restrict__ hhat,
                           const float* __restrict__ dinv, const float* __restrict__ b,
                           float* __restrict__ h, int N) {
    int i = blockIdx.x * blockDim.x + threadIdx.x;
    if (i < N * HID) {
        int n = i >> 6, j = i & 63;
        h[i] = dinv[n] * (agg[i] + hhat[i]) + b[j];
    }
}

// segment-sum pooling: out[map[n]][j] += in[n][j]
__global__ void k_pool(const int* __restrict__ map, const float* __restrict__ in,
                       float* out, int N) {
    int i = blockIdx.x * blockDim.x + threadIdx.x;
    if (i < N * HID) {
        int n = i >> 6, j = i & 63;
        atomicAdd(&out[(size_t)map[n] * HID + j], in[i]);
    }
}

__global__ void k_relu_bias(float* __restrict__ t, const float* __restrict__ b, int n) {
    int i = blockIdx.x * blockDim.x + threadIdx.x;
    if (i < n) t[i] = fmaxf(t[i] + b[i & 63], 0.0f);
}

// final 64x8 projection + log_softmax; one thread per graph row
__global__ void k_head2(const float* __restrict__ h1, const float* __restrict__ w2,
                        const float* __restrict__ b2, float* __restrict__ out, int G) {
    int r = threadIdx.x;
    if (r >= G) return;
    float acc[8];
#pragma unroll
    for (int o = 0; o < 8; ++o) acc[o] = b2[o];
    for (int j = 0; j < HID; ++j) {
        float hv = h1[(size_t)r * HID + j];
#pragma unroll
        for (int o = 0; o < 8; ++o) acc[o] += hv * w2[j * 8 + o];
    }
    float m = acc[0];
#pragma unroll
    for (int o = 1; o < 8; ++o) m = fmaxf(m, acc[o]);
    float s = 0.0f;
#pragma unroll
    for (int o = 0; o < 8; ++o) s += expf(acc[o] - m);
    float lse = logf(s);
#pragma unroll
    for (int o = 0; o < 8; ++o) out[r * 8 + o] = acc[o] - m - lse;
}

// ---------------- WMMA f32 GEMM: C[M x 64] = A[M x 64] @ B[64 x 64] ----------------
// One wave per 16-row tile; 4 n-tiles of 16 cols; K stepped by 4 through
// V_WMMA_F32_16X16X4_F32. Full f32 precision (no downconversion needed: the
// dense matmuls are ~0.1% of the memory-bound edge-aggregation cost).
__global__ void __launch_bounds__(256)
k_gemm64(const float* __restrict__ A, const float* __restrict__ B,
         float* __restrict__ C, int M) {
    const int lane = threadIdx.x & 31;
    const int mtile = blockIdx.x * (blockDim.x >> 5) + (threadIdx.x >> 5);
    if (mtile * 16 >= M) return;           // wave-uniform guard (EXEC stays all-1s)
    const int lo = lane & 15;
    const int hi = lane >> 4;
    const int row0 = mtile * 16;

    v8f acc[4] = {v8f{}, v8f{}, v8f{}, v8f{}};
    const float* arow = A + (size_t)(row0 + lo) * HID;
#pragma unroll
    for (int kb = 0; kb < HID; kb += 4) {
        // A fragment: lane lo = row, K = {kb+2*hi, kb+2*hi+1} (8B-aligned pair)
        const v2f a = *(const v2f*)(arow + kb + 2 * hi);
#pragma unroll
        for (int nt = 0; nt < 4; ++nt) {
            v2f b;
            b.x = B[(size_t)(kb + 2 * hi + 0) * HID + nt * 16 + lo];
            b.y = B[(size_t)(kb + 2 * hi + 1) * HID + nt * 16 + lo];
            acc[nt] = __builtin_amdgcn_wmma_f32_16x16x4_f32(
                false, a, false, b, (short)0, acc[nt], false, false);
        }
    }
#pragma unroll
    for (int nt = 0; nt < 4; ++nt) {
#pragma unroll
        for (int i = 0; i < 8; ++i) {
            C[(size_t)(row0 + i + 8 * hi) * HID + nt * 16 + lo] = acc[nt][i];
        }
    }
}

// ---------------- host orchestration ----------------

static inline int cdiv_ll(long long a, long long b) { return (int)((a + b - 1) / b); }
static inline size_t alignup(size_t v) { return (v + 255) & ~(size_t)255; }

extern "C" void kernel_launch(void* const* d_in, const int* in_sizes, int n_in,
                              void* d_out, int out_size, void* d_ws, size_t ws_size,
                              hipStream_t stream) {
    const float* x      = (const float*)d_in[0];
    const int*   ei     = (const int*)  d_in[1];   // (2,E): row0 = src, row1 = dst
    const int*   n2s    = (const int*)  d_in[2];
    const int*   s2g    = (const int*)  d_in[3];
    const float* W1     = (const float*)d_in[7];
    const float* b1     = (const float*)d_in[8];
    const float* Ws     = (const float*)d_in[9];
    const float* bs     = (const float*)d_in[10];
    const float* lin1_w = (const float*)d_in[11];
    const float* lin1_b = (const float*)d_in[12];
    const float* lin2_w = (const float*)d_in[13];
    const float* lin2_b = (const float*)d_in[14];
    float* out = (float*)d_out;

    const int N = in_sizes[0];          // x is (N,1)
    const int E = in_sizes[1] / 2;
    const int S = in_sizes[3];
    const int G = out_size / 8;

    // workspace carve-up
    char* p = (char*)d_ws;
    float* dinv = (float*)p; p += alignup((size_t)N * 4);
    float* h    = (float*)p; p += alignup((size_t)N * HID * 4);
    float* hhat = (float*)p; p += alignup((size_t)N * HID * 4);
    float* agg  = (float*)p; p += alignup((size_t)N * HID * 4);  // doubles as matmul temp
    float* sg   = (float*)p; p += alignup((size_t)S * HID * 4);
    float* g    = (float*)p; p += alignup((size_t)G * HID * 4);
    float* tg   = (float*)p; p += alignup((size_t)G * HID * 4);
    (void)ws_size;

    const int NE = N * HID;
    const int srcE = 0;  // ei + 0
    const int* src = ei;
    const int* dst = ei + E;

    // degrees (incl. self loop) -> dinv = rsqrt(deg)
    k_fill<<<cdiv_ll(N, 256), 256, 0, stream>>>(dinv, 1.0f, N);
    k_deg<<<cdiv_ll(E, 256), 256, 0, stream>>>(dst, dinv, E);
    k_rsqrt<<<cdiv_ll(N, 256), 256, 0, stream>>>(dinv, N);
    (void)srcE;

    const int scatterBlocks = cdiv_ll((long long)E * 32, 256);
    const int gemmBlocksN   = cdiv_ll(cdiv_ll(N, 16), 8);   // 8 waves (16-row tiles) per 256-thr block

    // ---- layer 1 (K=1 input, broadcast matmul) ----
    k_layer1<<<cdiv_ll(NE, 256), 256, 0, stream>>>(x, W1, agg, N);
    k_hhat_zero<<<cdiv_ll(NE, 256), 256, 0, stream>>>(agg, dinv, hhat, N);
    k_scatter<<<scatterBlocks, 256, 0, stream>>>(src, dst, hhat, agg, E);
    k_finalize<<<cdiv_ll(NE, 256), 256, 0, stream>>>(agg, hhat, dinv, b1, h, N);

    // ---- inner layers 2..4 (WMMA f32 GEMM) ----
    for (int l = 0; l < 3; ++l) {
        k_gemm64<<<gemmBlocksN, 256, 0, stream>>>(h, Ws + (size_t)l * HID * HID, agg, N);
        k_hhat_zero<<<cdiv_ll(NE, 256), 256, 0, stream>>>(agg, dinv, hhat, N);
        k_scatter<<<scatterBlocks, 256, 0, stream>>>(src, dst, hhat, agg, E);
        k_finalize<<<cdiv_ll(NE, 256), 256, 0, stream>>>(agg, hhat, dinv, bs + (size_t)l * HID, h, N);
    }

    // ---- two-stage global_add_pool ----
    k_fill<<<cdiv_ll((long long)S * HID, 256), 256, 0, stream>>>(sg, 0.0f, S * HID);
    k_fill<<<cdiv_ll((long long)G * HID, 256), 256, 0, stream>>>(g, 0.0f, G * HID);
    k_pool<<<cdiv_ll(NE, 256), 256, 0, stream>>>(n2s, h, sg, N);
    k_pool<<<cdiv_ll((long long)S * HID, 256), 256, 0, stream>>>(s2g, sg, g, S);

    // ---- head: relu(g @ lin1_w + lin1_b) @ lin2_w + lin2_b, log_softmax ----
    k_gemm64<<<cdiv_ll(cdiv_ll(G, 16), 8), 256, 0, stream>>>(g, lin1_w, tg, G);
    k_relu_bias<<<cdiv_ll((long long)G * HID, 256), 256, 0, stream>>>(tg, lin1_b, G * HID);
    k_head2<<<1, 64, 0, stream>>>(tg, lin2_w, lin2_b, out, G);
}